// GCN_model_9698036155052
// MI455X (gfx1250) — compile-verified
//
#include <hip/hip_runtime.h>
#include <math.h>

typedef __attribute__((ext_vector_type(2))) float v2f;
typedef __attribute__((ext_vector_type(8))) float v8f;

#define N_NODES 100000
#define D_IN    128
#define H1      16
#define H2      32
#define N_GRAPHS 64
#define N_GFEAT  7
#define N_CLASSES 4

// ---------------- degree / normalization ----------------

__global__ void k_degree(const int* __restrict__ dst, int E, float* __restrict__ deg) {
    int i = blockIdx.x * blockDim.x + threadIdx.x;
    if (i < E) unsafeAtomicAdd(&deg[dst[i]], 1.0f);
}

__global__ void k_dis(float* __restrict__ deg, int N) {
    int i = blockIdx.x * blockDim.x + threadIdx.x;
    if (i < N) {
        float d = deg[i] + 1.0f;           // +1 for self-loop; always > 0
        deg[i] = rsqrtf(d);                // deg[] becomes dis[]
    }
}

// ---------------- GEMM1: t1[N,16] = x[N,128] @ W1[128,16] (fp32 WMMA) ----------------

__global__ void k_gemm1(const float* __restrict__ x, const float* __restrict__ W1,
                        float* __restrict__ t1, int N) {
    int wave = blockIdx.x * (blockDim.x >> 5) + (threadIdx.x >> 5);
    int lane = threadIdx.x & 31;
    int row0 = wave * 16;
    if (row0 >= N) return;                 // wave-uniform exit: EXEC all-1s for WMMA
    int m     = lane & 15;                 // matrix row (A) / col (B,C)
    int khalf = lane >> 4;                 // 0: K=0,1  1: K=2,3 within a 4-wide K step
    int row   = row0 + m;

    v8f c = {};
    #pragma unroll
    for (int k0 = 0; k0 < D_IN; k0 += 4) {
        int kb = k0 + 2 * khalf;
        v2f a, b;
        a.x = x[row * D_IN + kb];
        a.y = x[row * D_IN + kb + 1];
        b.x = W1[kb * H1 + m];
        b.y = W1[(kb + 1) * H1 + m];
        c = __builtin_amdgcn_wmma_f32_16x16x4_f32(false, a, false, b,
                                                  (short)0, c, false, false);
    }
    #pragma unroll
    for (int r = 0; r < 8; ++r)
        t1[(row0 + r + 8 * khalf) * H1 + m] = c[r];
}

// ---------------- edge scatter layer 1: 16 lanes per edge ----------------

__global__ void k_scatter16(const int* __restrict__ src, const int* __restrict__ dst,
                            const float* __restrict__ dis, const float* __restrict__ t1,
                            float* __restrict__ agg1, int E) {
    int gid = blockIdx.x * blockDim.x + threadIdx.x;
    int e = gid >> 4;
    int f = gid & 15;
    if (e >= E) return;
    int s = src[e], d = dst[e];
    float nrm = dis[s] * dis[d];
    float v = nrm * t1[s * H1 + f];
    unsafeAtomicAdd(&agg1[d * H1 + f], v);
}

// ---------------- h1 = relu(agg1 + dis^2 * t1 + b1), in place in agg1 ----------------

__global__ void k_relu1(const float* __restrict__ t1, float* __restrict__ agg1,
                        const float* __restrict__ dis, const float* __restrict__ b1, int N) {
    int i = blockIdx.x * blockDim.x + threadIdx.x;
    if (i >= N * H1) return;
    int n = i >> 4, f = i & 15;
    float dn = dis[n];
    float v = agg1[i] + dn * dn * t1[i] + b1[f];
    agg1[i] = v > 0.0f ? v : 0.0f;
}

// ---------------- GEMM2: t2[N,32] = h1[N,16] @ W2[16,32] (fp32 WMMA, 2 N-tiles) ----------------

__global__ void k_gemm2(const float* __restrict__ h1, const float* __restrict__ W2,
                        float* __restrict__ t2, int N) {
    int wave = blockIdx.x * (blockDim.x >> 5) + (threadIdx.x >> 5);
    int lane = threadIdx.x & 31;
    int row0 = wave * 16;
    if (row0 >= N) return;
    int m     = lane & 15;
    int khalf = lane >> 4;
    int row   = row0 + m;

    v8f c0 = {}, c1 = {};
    #pragma unroll
    for (int k0 = 0; k0 < H1; k0 += 4) {
        int kb = k0 + 2 * khalf;
        v2f a, b0, b1v;
        a.x  = h1[row * H1 + kb];
        a.y  = h1[row * H1 + kb + 1];
        b0.x = W2[kb * H2 + m];
        b0.y = W2[(kb + 1) * H2 + m];
        b1v.x = W2[kb * H2 + 16 + m];
        b1v.y = W2[(kb + 1) * H2 + 16 + m];
        c0 = __builtin_amdgcn_wmma_f32_16x16x4_f32(false, a, false, b0,
                                                   (short)0, c0, false, false);
        c1 = __builtin_amdgcn_wmma_f32_16x16x4_f32(false, a, false, b1v,
                                                   (short)0, c1, false, false);
    }
    #pragma unroll
    for (int r = 0; r < 8; ++r) {
        int rr = row0 + r + 8 * khalf;
        t2[rr * H2 + m]      = c0[r];
        t2[rr * H2 + 16 + m] = c1[r];
    }
}

// ---------------- edge scatter layer 2: 32 lanes (one wave) per edge ----------------

__global__ void k_scatter32(const int* __restrict__ src, const int* __restrict__ dst,
                            const float* __restrict__ dis, const float* __restrict__ t2,
                            float* __restrict__ agg2, int E) {
    int gid = blockIdx.x * blockDim.x + threadIdx.x;
    int e = gid >> 5;
    int f = gid & 31;
    if (e >= E) return;
    int s = src[e], d = dst[e];
    float nrm = dis[s] * dis[d];
    float v = nrm * t2[s * H2 + f];
    unsafeAtomicAdd(&agg2[d * H2 + f], v);
}

// ---------------- h2 = relu(agg2 + dis^2 * t2 + b2), fused mean-pool accumulation ----------------

__global__ void k_relu2_pool(const float* __restrict__ t2, const float* __restrict__ agg2,
                             const float* __restrict__ dis, const float* __restrict__ b2,
                             const int* __restrict__ batch,
                             float* __restrict__ sums, float* __restrict__ cnt, int N) {
    int i = blockIdx.x * blockDim.x + threadIdx.x;
    if (i >= N * H2) return;
    int n = i >> 5, f = i & 31;          // one wave == one node, 32 distinct features
    float dn = dis[n];
    float v = agg2[i] + dn * dn * t2[i] + b2[f];
    v = v > 0.0f ? v : 0.0f;
    int g = batch[n];
    unsafeAtomicAdd(&sums[g * H2 + f], v);
    if (f == 0) unsafeAtomicAdd(&cnt[g], 1.0f);
}

// ---------------- final head: z = [pooled | gfeat] @ Wfc + bfc, log_softmax ----------------

__global__ void k_final(const float* __restrict__ sums, const float* __restrict__ cnt,
                        const float* __restrict__ gf, const float* __restrict__ Wfc,
                        const float* __restrict__ bfc, float* __restrict__ out, int G) {
    int g = blockIdx.x * blockDim.x + threadIdx.x;
    if (g >= G) return;
    float inv = 1.0f / fmaxf(cnt[g], 1.0f);
    float z[N_CLASSES];
    #pragma unroll
    for (int c = 0; c < N_CLASSES; ++c) {
        float acc = bfc[c];
        for (int j = 0; j < H2; ++j)
            acc += sums[g * H2 + j] * inv * Wfc[j * N_CLASSES + c];
        for (int k = 0; k < N_GFEAT; ++k)
            acc += gf[g * N_GFEAT + k] * Wfc[(H2 + k) * N_CLASSES + c];
        z[c] = acc;
    }
    float m = fmaxf(fmaxf(z[0], z[1]), fmaxf(z[2], z[3]));
    float s = 0.0f;
    #pragma unroll
    for (int c = 0; c < N_CLASSES; ++c) s += __expf(z[c] - m);
    float lse = m + __logf(s);
    #pragma unroll
    for (int c = 0; c < N_CLASSES; ++c) out[g * N_CLASSES + c] = z[c] - lse;
}

// ---------------- host-side launcher ----------------

extern "C" void kernel_launch(void* const* d_in, const int* in_sizes, int n_in,
                              void* d_out, int out_size, void* d_ws, size_t ws_size,
                              hipStream_t stream) {
    const float* x   = (const float*)d_in[0];
    const int*   ei  = (const int*)  d_in[1];
    const int*   bat = (const int*)  d_in[2];
    const float* gf  = (const float*)d_in[3];
    const float* W1  = (const float*)d_in[4];
    const float* b1  = (const float*)d_in[5];
    const float* W2  = (const float*)d_in[6];
    const float* b2  = (const float*)d_in[7];
    const float* Wfc = (const float*)d_in[8];
    const float* bfc = (const float*)d_in[9];

    const int N = in_sizes[0] / D_IN;        // 100000
    const int E = in_sizes[1] / 2;           // 6400000
    const int G = in_sizes[3] / N_GFEAT;     // 64
    const int* src = ei;
    const int* dst = ei + E;

    // workspace layout (floats)
    float* ws   = (float*)d_ws;
    float* deg  = ws;                        // N      (becomes dis[])
    float* t1   = deg  + N;                  // N*H1
    float* agg1 = t1   + (size_t)N * H1;     // N*H1   (becomes h1 in place)
    float* t2   = agg1 + (size_t)N * H1;     // N*H2
    float* agg2 = t2   + (size_t)N * H2;     // N*H2
    float* sums = agg2 + (size_t)N * H2;     // G*H2
    float* cnt  = sums + (size_t)G * H2;     // G

    // zero all accumulators every call (graph-replay safe)
    hipMemsetAsync(deg,  0, (size_t)N * sizeof(float), stream);
    hipMemsetAsync(agg1, 0, (size_t)N * H1 * sizeof(float), stream);
    hipMemsetAsync(agg2, 0, (size_t)N * H2 * sizeof(float), stream);
    hipMemsetAsync(sums, 0, ((size_t)G * H2 + G) * sizeof(float), stream);

    const int BS = 256;

    k_degree<<<(E + BS - 1) / BS, BS, 0, stream>>>(dst, E, deg);
    k_dis<<<(N + BS - 1) / BS, BS, 0, stream>>>(deg, N);

    int tiles = (N + 15) / 16;               // 6250 waves, 8 waves per 256-thread block
    int wpb = BS / 32;
    k_gemm1<<<(tiles + wpb - 1) / wpb, BS, 0, stream>>>(x, W1, t1, N);

    long long th1 = (long long)E * H1;
    k_scatter16<<<(unsigned)((th1 + BS - 1) / BS), BS, 0, stream>>>(src, dst, deg, t1, agg1, E);
    k_relu1<<<(N * H1 + BS - 1) / BS, BS, 0, stream>>>(t1, agg1, deg, b1, N);

    k_gemm2<<<(tiles + wpb - 1) / wpb, BS, 0, stream>>>(agg1, W2, t2, N);

    long long th2 = (long long)E * H2;
    k_scatter32<<<(unsigned)((th2 + BS - 1) / BS), BS, 0, stream>>>(src, dst, deg, t2, agg2, E);
    k_relu2_pool<<<(N * H2 + BS - 1) / BS, BS, 0, stream>>>(t2, agg2, deg, b2, bat, sums, cnt, N);

    k_final<<<(G + 63) / 64, 64, 0, stream>>>(sums, cnt, gf, Wfc, bfc, (float*)d_out, G);
}